// MP_Block_46918222742293
// MI455X (gfx1250) — compile-verified
//
#include <hip/hip_runtime.h>

typedef float v2f __attribute__((ext_vector_type(2)));
typedef float v8f __attribute__((ext_vector_type(8)));

constexpr int NN   = 100000;   // nodes
constexpr int NE   = 1600000;  // edges
constexpr int FIN  = 32;
constexpr int FHID = 64;

// ---------------- utility kernels ----------------

__global__ void fill_kernel(float* __restrict__ p, float v, int n) {
  int i = blockIdx.x * blockDim.x + threadIdx.x;
  if (i < n) p[i] = v;
}

__global__ void copy_kernel(const float* __restrict__ src, float* __restrict__ dst, int n) {
  int i = blockIdx.x * blockDim.x + threadIdx.x;
  if (i < n) dst[i] = src[i];
}

// deg[col] += ew  (deg pre-filled with 1.0 == self-loop weight)
__global__ void degree_kernel(const int* __restrict__ ei, const float* __restrict__ ew,
                              float* __restrict__ deg, int nedges) {
  int e = blockIdx.x * blockDim.x + threadIdx.x;
  if (e < nedges) atomicAdd(&deg[ei[nedges + e]], ew[e]);
}

// deg -> dinv in place
__global__ void rsqrt_kernel(float* __restrict__ deg, int n) {
  int i = blockIdx.x * blockDim.x + threadIdx.x;
  if (i < n) {
    float d = deg[i];
    deg[i] = d > 0.0f ? rsqrtf(d) : 0.0f;
  }
}

// ---------------- fp32 WMMA GEMM: C[M,N] = A[M,K] @ B[K,N] ----------------
// one wave per 16x16 C tile; K multiple of 4; M multiple of 16; N multiple of 16.
__global__ void __launch_bounds__(256)
gemm_wmma_f32_kernel(const float* __restrict__ A, const float* __restrict__ B,
                     float* __restrict__ C, int M, int N, int K,
                     int tilesTotal, int tilesN) {
  int wave = (int)((blockIdx.x * blockDim.x + threadIdx.x) >> 5);
  if (wave >= tilesTotal) return;            // wave-uniform: EXEC stays all-1s
  int lane = threadIdx.x & 31;
  int tm = wave / tilesN;
  int tn = wave - tm * tilesN;
  int half = lane >> 4;                      // 0: lanes 0-15, 1: lanes 16-31
  int l    = lane & 15;

  // A 16x4 step layout: lanes 0-15 hold K = {k0, k0+1}; lanes 16-31 hold {k0+2, k0+3}
  const float* arow = A + (size_t)(tm * 16 + l) * K + half * 2;
  // B 4x16 step layout mirrors A: row k0+half*2 (+1), cols striped over 16 lanes
  const float* bcol = B + (size_t)(half * 2) * N + tn * 16 + l;

  v8f acc = {};
  for (int k0 = 0; k0 < K; k0 += 4) {
    v2f a, b;
    a.x = arow[k0];
    a.y = arow[k0 + 1];
    b.x = bcol[(size_t)k0 * N];
    b.y = bcol[(size_t)(k0 + 1) * N];
    acc = __builtin_amdgcn_wmma_f32_16x16x4_f32(
        /*neg_a=*/false, a, /*neg_b=*/false, b,
        /*c_mod=*/(short)0, acc, /*reuse_a=*/false, /*reuse_b=*/false);
  }

  // C/D layout: VGPR r -> row tm*16 + r + half*8, col tn*16 + l
  float* crow = C + (size_t)(tm * 16 + half * 8) * N + tn * 16 + l;
#pragma unroll
  for (int r = 0; r < 8; ++r) crow[(size_t)r * N] = acc[r];
}

// ---------------- edge scatter: out[col] += dinv[row]*ew*dinv[col] * xw[row] --------
// one thread per (edge, 4-feature group); lanes cover an edge's row contiguously.
__global__ void __launch_bounds__(256)
edge_scatter_kernel(const int* __restrict__ ei, const float* __restrict__ ew,
                    const float* __restrict__ dinv, const float* __restrict__ xw,
                    float* __restrict__ out, int nedges, int gshift, int F) {
  int gid = blockIdx.x * blockDim.x + threadIdx.x;
  int e = gid >> gshift;
  if (e >= nedges) return;
  int g = gid & ((1 << gshift) - 1);

  int row = ei[e];
  int col = ei[nedges + e];
  float norm = dinv[row] * ew[e] * dinv[col];

  const float4 v = *((const float4*)(xw + (size_t)row * F) + g);
  float* dst = out + (size_t)col * F + (g << 2);
  atomicAdd(dst + 0, norm * v.x);
  atomicAdd(dst + 1, norm * v.y);
  atomicAdd(dst + 2, norm * v.z);
  atomicAdd(dst + 3, norm * v.w);
}

// ---------------- epilogue: agg = relu(agg + dinv[i]^2 * xw + bias) ----------------
__global__ void bias_relu_self_kernel(float* __restrict__ agg, const float* __restrict__ xw,
                                      const float* __restrict__ dinv,
                                      const float* __restrict__ bias,
                                      int total, int fshift) {
  int gid = blockIdx.x * blockDim.x + threadIdx.x;
  if (gid >= total) return;
  int i = gid >> fshift;
  int f = gid & ((1 << fshift) - 1);
  float di = dinv[i];
  float v = agg[gid] + di * di * xw[gid] + bias[f];
  agg[gid] = v > 0.0f ? v : 0.0f;
}

// ---------------- launcher ----------------

extern "C" void kernel_launch(void* const* d_in, const int* in_sizes, int n_in,
                              void* d_out, int out_size, void* d_ws, size_t ws_size,
                              hipStream_t stream) {
  const float* x  = (const float*)d_in[0];   // [NN, FIN]
  const int*   ei = (const int*)d_in[1];     // [2, NE] flat
  const float* ea = (const float*)d_in[2];   // [NE, 1]
  const float* W1 = (const float*)d_in[3];   // [FIN, FHID]
  const float* b1 = (const float*)d_in[4];   // [FHID]
  const float* W2 = (const float*)d_in[5];   // [FHID, FIN]
  const float* b2 = (const float*)d_in[6];   // [FIN]

  float* out = (float*)d_out;                // [NN*FIN] ++ [NE]
  float* ws  = (float*)d_ws;

  float* dinv = ws;                          // NN floats (deg -> dinv)
  float* buf1 = ws + 102400;                 // NN*FHID floats (xw1, later xw2)
  float* buf2 = buf1 + (size_t)NN * FHID;    // NN*FHID floats (agg1 -> h)

  const int B = 256;
  auto blocks = [](long long n, int b) { return (int)((n + b - 1) / b); };

  // degrees (self-loop weight folded in as init value 1.0)
  fill_kernel<<<blocks(NN, B), B, 0, stream>>>(dinv, 1.0f, NN);
  degree_kernel<<<blocks(NE, B), B, 0, stream>>>(ei, ea, dinv, NE);
  rsqrt_kernel<<<blocks(NN, B), B, 0, stream>>>(dinv, NN);

  // ---- layer 1 ----
  {
    int tilesN = FHID / 16, tilesTotal = (NN / 16) * tilesN;
    gemm_wmma_f32_kernel<<<blocks((long long)tilesTotal * 32, B), B, 0, stream>>>(
        x, W1, buf1, NN, FHID, FIN, tilesTotal, tilesN);
  }
  fill_kernel<<<blocks((long long)NN * FHID, B), B, 0, stream>>>(buf2, 0.0f, NN * FHID);
  edge_scatter_kernel<<<blocks((long long)NE * (FHID / 4), B), B, 0, stream>>>(
      ei, ea, dinv, buf1, buf2, NE, /*gshift=*/4, FHID);
  bias_relu_self_kernel<<<blocks((long long)NN * FHID, B), B, 0, stream>>>(
      buf2, buf1, dinv, b1, NN * FHID, /*fshift=*/6);

  // ---- layer 2 ----
  {
    int tilesN = FIN / 16, tilesTotal = (NN / 16) * tilesN;
    gemm_wmma_f32_kernel<<<blocks((long long)tilesTotal * 32, B), B, 0, stream>>>(
        buf2, W2, buf1, NN, FIN, FHID, tilesTotal, tilesN);
  }
  fill_kernel<<<blocks((long long)NN * FIN, B), B, 0, stream>>>(out, 0.0f, NN * FIN);
  edge_scatter_kernel<<<blocks((long long)NE * (FIN / 4), B), B, 0, stream>>>(
      ei, ea, dinv, buf1, out, NE, /*gshift=*/3, FIN);
  bias_relu_self_kernel<<<blocks((long long)NN * FIN, B), B, 0, stream>>>(
      out, buf1, dinv, b2, NN * FIN, /*fshift=*/5);

  // second tuple element: edge_attr passthrough
  copy_kernel<<<blocks(NE, B), B, 0, stream>>>(ea, out + (size_t)NN * FIN, NE);
}